// recon_6face_loss_23785528885856
// MI455X (gfx1250) — compile-verified
//
#include <hip/hip_runtime.h>
#include <hip/hip_bf16.h>

// Problem constants (from reference): B=128, N=8192, 6 faces.
#define B_      128
#define N_      8192
#define THREADS 512
#define WAVES   16          // 512 threads / wave32

typedef __attribute__((ext_vector_type(2))) float v2f;   // A/B operand of WMMA f32 16x16x4 (2 VGPRs)
typedef __attribute__((ext_vector_type(8))) float v8f;   // C/D operand (8 VGPRs)

// -----------------------------------------------------------------------------
// Kernel 1: one workgroup per batch b. Streams pc/face_shift/face_log_var/mask
// contiguously exactly once (memory-bound: ~118MB total -> ~5us at 23.3 TB/s).
// Two-pass masked softmax + 10 weighted moments per face:
//   Z = sum u,  s1 = sum u*P,  S2 = sum u*P Pt   with u = exp(min_std - std)
// Final cross-wave reduction of the 16 wave-partials is done with chained
// V_WMMA_F32_16X16X4_F32: A[m][k] = wave-k partial of moment m, B = ones,
// D accumulated across 4 WMMAs covering the 16 waves (exact f32 math).
// -----------------------------------------------------------------------------
__global__ __launch_bounds__(THREADS)
void moments_kernel(const float* __restrict__ pc,
                    const float* __restrict__ fsh,
                    const float* __restrict__ flv,
                    const int*   __restrict__ pm,
                    float* __restrict__ mom)
{
    __shared__ float WPmin[6][WAVES];
    __shared__ float WP[6][WAVES][16];     // [face][wave][moment]

    const int tid  = threadIdx.x;
    const int lane = tid & 31;
    const int wav  = tid >> 5;
    const int b    = blockIdx.x;
    const long long bN = (long long)b * N_;

    // ---------------- pass 1: per-face min of masked std ----------------
    float mn[6];
#pragma unroll
    for (int f = 0; f < 6; ++f) mn[f] = 3.4e38f;

    for (int n = tid; n < N_; n += THREADS) {
        const int msk = pm[bN + n];
        const float* lv = flv + (bN + n) * 6;
#pragma unroll
        for (int f = 0; f < 6; ++f) {
            float s = (msk > 0) ? __expf(0.5f * lv[f]) : 1.0e10f;  // MASKED_STD
            mn[f] = fminf(mn[f], s);
        }
    }
#pragma unroll
    for (int f = 0; f < 6; ++f) {
#pragma unroll
        for (int off = 16; off >= 1; off >>= 1)
            mn[f] = fminf(mn[f], __shfl_xor(mn[f], off));
        if (lane == 0) WPmin[f][wav] = mn[f];
    }
    __syncthreads();
    float mnf[6];
#pragma unroll
    for (int f = 0; f < 6; ++f) {
        float m = WPmin[f][0];
#pragma unroll
        for (int w = 1; w < WAVES; ++w) m = fminf(m, WPmin[f][w]);
        mnf[f] = m;
    }

    // ---------------- pass 2: weighted moments ----------------
    float acc[6][10];
#pragma unroll
    for (int f = 0; f < 6; ++f)
#pragma unroll
        for (int m = 0; m < 10; ++m) acc[f][m] = 0.0f;

    for (int n = tid; n < N_; n += THREADS) {
        const int msk   = pm[bN + n];
        const float* lv = flv + (bN + n) * 6;
        const float* sh = fsh + (bN + n) * 18;
        const float* p  = pc  + (bN + n) * 3;
        if (n + THREADS < N_) {           // prefetch next tile of the big streams
            __builtin_prefetch(sh + THREADS * 18, 0, 0);
            __builtin_prefetch(p  + THREADS * 3,  0, 0);
        }
        const float px = p[0], py = p[1], pz = p[2];
#pragma unroll
        for (int f = 0; f < 6; ++f) {
            float s = (msk > 0) ? __expf(0.5f * lv[f]) : 1.0e10f;
            float u = __expf(mnf[f] - s);          // masked -> exp(-1e10)=0; all-masked -> 1
            float x = px + sh[f * 3 + 0];
            float y = py + sh[f * 3 + 1];
            float z = pz + sh[f * 3 + 2];
            float ux = u * x, uy = u * y, uz = u * z;
            acc[f][0] += u;
            acc[f][1] += ux;       acc[f][2] += uy;       acc[f][3] += uz;
            acc[f][4] += ux * x;   acc[f][5] += ux * y;   acc[f][6] += ux * z;
            acc[f][7] += uy * y;   acc[f][8] += uy * z;   acc[f][9] += uz * z;
        }
    }

    // intra-wave reduce, wave leader deposits partials in LDS
#pragma unroll
    for (int f = 0; f < 6; ++f) {
#pragma unroll
        for (int m = 0; m < 10; ++m) {
            float v = acc[f][m];
#pragma unroll
            for (int off = 16; off >= 1; off >>= 1)
                v += __shfl_xor(v, off);
            if (lane == 0) WP[f][wav][m] = v;
        }
    }
    __syncthreads();

    // ---------------- WMMA cross-wave reduction (wave 0, EXEC all ones) -------
    if (tid < 32) {
        const int row = lane & 15;        // = M index (moment)
        const int hi  = lane >> 4;        // lanes 16-31 supply K=2,3 per A layout
        const v2f bones = {1.0f, 1.0f};   // B = all-ones 4x16
#pragma unroll
        for (int f = 0; f < 6; ++f) {
            v8f d = {0.f, 0.f, 0.f, 0.f, 0.f, 0.f, 0.f, 0.f};
#pragma unroll
            for (int kk = 0; kk < 4; ++kk) {
                // A[m][k] = WP[f][kk*4 + k][m]; lane layout per ISA 16x4 f32 A-matrix
                v2f a = { WP[f][kk * 4 + hi * 2 + 0][row],
                          WP[f][kk * 4 + hi * 2 + 1][row] };
                d = __builtin_amdgcn_wmma_f32_16x16x4_f32(
                        /*neg_a=*/false, a, /*neg_b=*/false, bones,
                        /*c_mod=*/(short)0, d, /*reuse_a=*/false, /*reuse_b=*/false);
            }
            // D[m][n] = sum over 16 waves of moment m (broadcast over n).
            // C/D layout: VGPR r on lanes 0-15 -> M=r ; lanes 16-31 -> M=r+8.
            float* dst = mom + ((long long)b * 6 + f) * 16;
            if (lane == 0) {
#pragma unroll
                for (int r = 0; r < 8; ++r) dst[r] = d[r];
            }
            if (lane == 16) { dst[8] = d[0]; dst[9] = d[1]; }
        }
    }
}

// -----------------------------------------------------------------------------
// Kernel 2: one thread per (b,f) = 768 threads, single block. Closed-form 3x3
// symmetric eigensolve (trig method), plane residual vs GT, deterministic
// block reduction to the scalar loss. dn is invariant to eigenvector sign.
// -----------------------------------------------------------------------------
__global__ __launch_bounds__(768)
void finish_kernel(const float* __restrict__ mom,
                   const float* __restrict__ gtR,
                   const float* __restrict__ gtt,
                   const float* __restrict__ gts,
                   const int*   __restrict__ sym,
                   float* __restrict__ out)
{
    __shared__ float part[24];
    const int t = threadIdx.x;          // 0..767
    const int b = t / 6, f = t % 6;

    const float* M = mom + (long long)t * 16;
    float Z  = fmaxf(M[0], 1e-30f);     // Z >= 1 in practice (min achiever has u=1)
    float mx = M[1] / Z, my = M[2] / Z, mz = M[3] / Z;   // mu (wsum==1 for softmax)
    float cxx = M[4] / Z - mx * mx;
    float cxy = M[5] / Z - mx * my;
    float cxz = M[6] / Z - mx * mz;
    float cyy = M[7] / Z - my * my;
    float cyz = M[8] / Z - my * mz;
    float czz = M[9] / Z - mz * mz;

    // smallest eigenvalue of symmetric C (trigonometric method)
    float q   = (cxx + cyy + czz) * (1.0f / 3.0f);
    float p1  = cxy * cxy + cxz * cxz + cyz * cyz;
    float axx = cxx - q, ayy = cyy - q, azz = czz - q;
    float p2  = axx * axx + ayy * ayy + azz * azz + 2.0f * p1;
    float p   = sqrtf(p2 * (1.0f / 6.0f)) + 1e-30f;
    float ip  = 1.0f / p;
    float bxx = axx * ip, bxy = cxy * ip, bxz = cxz * ip;
    float byy = ayy * ip, byz = cyz * ip, bzz = azz * ip;
    float detB = bxx * (byy * bzz - byz * byz)
               - bxy * (bxy * bzz - byz * bxz)
               + bxz * (bxy * byz - byy * bxz);
    float r   = fminf(1.0f, fmaxf(-1.0f, 0.5f * detB));
    float phi = acosf(r) * (1.0f / 3.0f);
    float lam = q + 2.0f * p * cosf(phi + 2.0943951023931953f);  // smallest

    // eigenvector: largest cross product of rows of (C - lam I)
    float r0x = cxx - lam, r0y = cxy,       r0z = cxz;
    float r1x = cxy,       r1y = cyy - lam, r1z = cyz;
    float r2x = cxz,       r2y = cyz,       r2z = czz - lam;
    float c0x = r0y * r1z - r0z * r1y, c0y = r0z * r1x - r0x * r1z, c0z = r0x * r1y - r0y * r1x;
    float c1x = r0y * r2z - r0z * r2y, c1y = r0z * r2x - r0x * r2z, c1z = r0x * r2y - r0y * r2x;
    float c2x = r1y * r2z - r1z * r2y, c2y = r1z * r2x - r1x * r2z, c2z = r1x * r2y - r1y * r2x;
    float n0 = c0x * c0x + c0y * c0y + c0z * c0z;
    float n1 = c1x * c1x + c1y * c1y + c1z * c1z;
    float n2 = c2x * c2x + c2y * c2y + c2z * c2z;
    float nx = c0x, ny = c0y, nz = c0z, nn = n0;
    if (n1 > nn) { nx = c1x; ny = c1y; nz = c1z; nn = n1; }
    if (n2 > nn) { nx = c2x; ny = c2y; nz = c2z; nn = n2; }
    if (nn < 1e-30f) { nx = 1.0f; ny = 0.0f; nz = 0.0f; nn = 1.0f; }
    float inv = rsqrtf(nn);
    nx *= inv; ny *= inv; nz *= inv;

    float cc  = -(nx * mx + ny * my + nz * mz);
    float dnx = cc * nx, dny = cc * ny, dnz = cc * nz;

    // GT plane: axes=[1,0,2,0,2,1], signs=[1,1,1,-1,-1,-1]
    int   ax = (f == 0 || f == 5) ? 1 : ((f == 1 || f == 3) ? 0 : 2);
    float sg = (f < 3) ? 1.0f : -1.0f;
    float dx = sg * gtR[b * 9 + 0 + ax];
    float dy = sg * gtR[b * 9 + 3 + ax];
    float dz = sg * gtR[b * 9 + 6 + ax];
    float hs = 0.5f * gts[b * 3 + ax];
    float ppx = gtt[b * 3 + 0] + dx * hs;
    float ppy = gtt[b * 3 + 1] + dy * hs;
    float ppz = gtt[b * 3 + 2] + dz * hs;
    float cg = -(dx * ppx + dy * ppy + dz * ppz);
    float gx = dx * cg, gy = dy * cg, gz = dz * cg;

    float res  = (fabsf(dnx - gx) + fabsf(dny - gy) + fabsf(dnz - gz)) * (1.0f / 3.0f);
    float keep = (sym[b * 4] == 0) ? 1.0f : ((f == 0 || f == 5) ? 1.0f : 0.0f);
    float v    = res * keep * (1.0f / (float)B_);

#pragma unroll
    for (int off = 16; off >= 1; off >>= 1) v += __shfl_xor(v, off);
    if ((t & 31) == 0) part[t >> 5] = v;
    __syncthreads();
    if (t == 0) {
        float s = 0.0f;
#pragma unroll
        for (int i = 0; i < 24; ++i) s += part[i];
        out[0] = s;
    }
}

// -----------------------------------------------------------------------------
extern "C" void kernel_launch(void* const* d_in, const int* in_sizes, int n_in,
                              void* d_out, int out_size, void* d_ws, size_t ws_size,
                              hipStream_t stream)
{
    (void)in_sizes; (void)n_in; (void)out_size; (void)ws_size;
    const float* pc  = (const float*)d_in[0];   // (B,N,3)
    const float* fsh = (const float*)d_in[1];   // (B,N,18)
    const float* flv = (const float*)d_in[2];   // (B,N,6)
    const float* gtR = (const float*)d_in[3];   // (B,3,3)
    const float* gtt = (const float*)d_in[4];   // (B,3)
    const float* gts = (const float*)d_in[5];   // (B,3)
    const int*   sym = (const int*)d_in[6];     // (B,4)
    // d_in[7] = obj_ids : unused by the reference
    const int*   pm  = (const int*)d_in[8];     // (B,N)

    float* mom = (float*)d_ws;                  // 768 * 16 floats = 48 KB
    float* out = (float*)d_out;                 // scalar f32

    moments_kernel<<<B_, THREADS, 0, stream>>>(pc, fsh, flv, pm, mom);
    finish_kernel<<<1, 768, 0, stream>>>(mom, gtR, gtt, gts, sym, out);
}